// PFE_59760174957086
// MI455X (gfx1250) — compile-verified
//
#include <hip/hip_runtime.h>
#include <hip/hip_bf16.h>

// ---- problem constants (match reference) ----
#define PP    96000
#define MM    32
#define CINc  10
#define COUTc 64
#define BBc   8
#define NXc   432
#define NYc   496
#define NYNX  (NYc * NXc)          // 214272
#define BN_EPS 1e-3f

#define PXc 0.16f
#define PYc 0.16f
#define PZc 4.0f
#define XOFFc (PXc * 0.5f + 0.0f)     //  0.08
#define YOFFc (PYc * 0.5f - 39.68f)   // -39.60
#define ZOFFc (PZc * 0.5f - 3.0f)     // -1.00

#define NBLK2 ((PP + 255) / 256)      // 375

typedef __attribute__((ext_vector_type(2))) float v2f;
typedef __attribute__((ext_vector_type(8))) float v8f;

// ---------------------------------------------------------------------------
// Kernel 1: zero the 439MB BEV output (float4 grid-stride; HBM-bound step)
// ---------------------------------------------------------------------------
__global__ __launch_bounds__(256) void zero_out_k(float* __restrict__ p, long long n) {
    long long n4 = n >> 2;
    float4* p4 = (float4*)p;
    long long i = (long long)blockIdx.x * blockDim.x + threadIdx.x;
    long long stride = (long long)gridDim.x * blockDim.x;
    float4 z; z.x = 0.f; z.y = 0.f; z.z = 0.f; z.w = 0.f;
    for (; i < n4; i += stride) p4[i] = z;
    if (blockIdx.x == 0 && threadIdx.x == 0)
        for (long long t = n4 << 2; t < n; ++t) p[t] = 0.f;
}

// ---------------------------------------------------------------------------
// Kernel 2: per-pillar stats (cluster mean, voxel center, out base) +
// deterministic block-partial reduction of fbar[10] and S=E[f f^T] (55 tri).
// One thread per pillar; shuffle tree within wave, LDS across 8 waves.
// ---------------------------------------------------------------------------
__global__ __launch_bounds__(256) void pillar_stats_k(
    const float* __restrict__ pillars, const int* __restrict__ coords,
    const int* __restrict__ nump, float* __restrict__ stats,
    float* __restrict__ part) {
    __shared__ float sh[65 * 8];
    const int tid = threadIdx.x;
    const int p   = blockIdx.x * 256 + tid;

    float accF[10], accS[55];
#pragma unroll
    for (int i = 0; i < 10; ++i) accF[i] = 0.f;
#pragma unroll
    for (int i = 0; i < 55; ++i) accS[i] = 0.f;

    if (p < PP) {
        const float* pp = pillars + (size_t)p * MM * 4;
        float sx = 0.f, sy = 0.f, sz = 0.f;
#pragma unroll 4
        for (int m = 0; m < MM; ++m) {
            sx += pp[m * 4 + 0]; sy += pp[m * 4 + 1]; sz += pp[m * 4 + 2];
        }
        const float n  = (float)nump[p];
        const float mx = sx / n, my = sy / n, mz = sz / n;
        const int cb = coords[p * 4 + 0], cz = coords[p * 4 + 1];
        const int cy = coords[p * 4 + 2], cx = coords[p * 4 + 3];
        const float cenx = (float)cx * PXc + XOFFc;
        const float ceny = (float)cy * PYc + YOFFc;
        const float cenz = (float)cz * PZc + ZOFFc;
        float* st = stats + (size_t)p * 8;
        st[0] = mx; st[1] = my; st[2] = mz;
        st[3] = cenx; st[4] = ceny; st[5] = cenz;
        // flat BEV base: b*(COUT*NY*NX) + z + y*NX + x  (channel stride NYNX)
        ((int*)st)[6] = cb * (COUTc * NYNX) + cz + cy * NXc + cx;

        for (int m = 0; m < MM; ++m) {
            const float x = pp[m * 4 + 0], y = pp[m * 4 + 1];
            const float z = pp[m * 4 + 2], w = pp[m * 4 + 3];
            float f[10];
            f[0] = x; f[1] = y; f[2] = z; f[3] = w;
            f[4] = x - mx;   f[5] = y - my;   f[6] = z - mz;
            f[7] = x - cenx; f[8] = y - ceny; f[9] = z - cenz;
#pragma unroll
            for (int i = 0; i < 10; ++i) accF[i] += f[i];
            int idx = 0;
#pragma unroll
            for (int i = 0; i < 10; ++i)
#pragma unroll
                for (int j = i; j < 10; ++j) accS[idx++] += f[i] * f[j];
        }
    }

    const int lane = tid & 31, wid = tid >> 5;
    auto redw = [&](float v, int a) {
#pragma unroll
        for (int off = 16; off > 0; off >>= 1) v += __shfl_xor(v, off, 32);
        if (lane == 0) sh[a * 8 + wid] = v;
    };
#pragma unroll
    for (int i = 0; i < 10; ++i) redw(accF[i], i);
#pragma unroll
    for (int i = 0; i < 55; ++i) redw(accS[i], 10 + i);
    __syncthreads();
    if (tid < 65) {
        float t = 0.f;
#pragma unroll
        for (int w = 0; w < 8; ++w) t += sh[tid * 8 + w];
        part[blockIdx.x * 72 + tid] = t;   // fixed-order => deterministic
    }
}

// ---------------------------------------------------------------------------
// Kernel 3: fixed-order final reduce + BN coefficients.
//   mu_c = fbar . W_c ;  E[h^2]_c = W_c^T S W_c ;  var = E[h^2]-mu^2
//   scale = gamma*rsqrt(var+eps) ; shift = beta - mu*scale
// ---------------------------------------------------------------------------
__global__ __launch_bounds__(128) void bn_coeffs_k(
    const float* __restrict__ part, const float* __restrict__ W,
    const float* __restrict__ gamma, const float* __restrict__ beta,
    float* __restrict__ sscale) {
    __shared__ float Sf[100];
    __shared__ float fb[10];
    const int tid = threadIdx.x;
    if (tid < 65) {
        float t = 0.f;
        for (int b = 0; b < NBLK2; ++b) t += part[b * 72 + tid];
        t *= 1.0f / ((float)PP * (float)MM);
        if (tid < 10) {
            fb[tid] = t;
        } else {
            int r = tid - 10, i = 0;
            while (r >= 10 - i) { r -= 10 - i; ++i; }
            const int j = i + r;
            Sf[i * 10 + j] = t; Sf[j * 10 + i] = t;
        }
    }
    __syncthreads();
    if (tid < COUTc) {
        float wc[10];
#pragma unroll
        for (int i = 0; i < 10; ++i) wc[i] = W[i * COUTc + tid];
        float mu = 0.f;
#pragma unroll
        for (int i = 0; i < 10; ++i) mu += fb[i] * wc[i];
        float e2 = 0.f;
#pragma unroll
        for (int i = 0; i < 10; ++i) {
            float a = 0.f;
#pragma unroll
            for (int j = 0; j < 10; ++j) a += Sf[i * 10 + j] * wc[j];
            e2 += wc[i] * a;
        }
        const float var = e2 - mu * mu;
        const float inv = rsqrtf(var + BN_EPS);
        const float sc  = gamma[tid] * inv;
        sscale[tid]         = sc;
        sscale[COUTc + tid] = beta[tid] - mu * sc;
    }
}

// ---------------------------------------------------------------------------
// Kernel 4: WMMA GEMM + BN affine + ReLU + max-over-points + atomicMax scatter.
// One wave = 16 feat rows (half a pillar; 32 rows/pillar => tiles never split).
// A (16x4 f32): lanes 0-15 hold K{0,1} in v0/v1, lanes 16-31 hold K{2,3}.
// B (4x16 f32): mirrored — lane n=L&15, K = kc*4 + (L>>4)*2 + v.
// C (16x16 f32): lane L holds channel N=L&15, rows M = (L>>4)*8 + j.
// K padded 10 -> 12 with zeros. For the padded K-chunk (kc=2, kp=1) the A
// fragment is forced to zero, so the B fragment can unconditionally load the
// valid rows 8/9 — no divergent guards, no OOB, products are 0 either way.
// ---------------------------------------------------------------------------
__global__ __launch_bounds__(256) void pfe_wmma_k(
    const float* __restrict__ pillars, const float* __restrict__ W,
    const float* __restrict__ stats, const float* __restrict__ sscale,
    float* __restrict__ out) {
    const int tid  = threadIdx.x;
    const int lane = tid & 31;
    const int wid  = tid >> 5;
    const int g    = blockIdx.x * 8 + wid;   // global wave index
    const int p    = g >> 1;                 // pillar
    const int m0   = (g & 1) << 4;           // 0 or 16: which half of pillar
    const int mrow = lane & 15;
    const bool hi  = (lane >> 4) != 0;       // kp: K-pair select
    const int m    = m0 + mrow;

    const float* pt = pillars + ((size_t)p * MM + m) * 4;
    const float x = pt[0], y = pt[1], z = pt[2], w = pt[3];
    const float* st = stats + (size_t)p * 8;
    const float f0 = x, f1 = y, f2 = z, f3 = w;
    const float f4 = x - st[0], f5 = y - st[1], f6 = z - st[2];
    const float f7 = x - st[3], f8 = y - st[4], f9 = z - st[5];
    const int obase = ((const int*)st)[6];

    // A fragments: 6 plain cndmasks, no dynamic array indexing.
    v2f a0, a1, a2;
    a0[0] = hi ? f2 : f0;  a0[1] = hi ? f3 : f1;
    a1[0] = hi ? f6 : f4;  a1[1] = hi ? f7 : f5;
    a2[0] = hi ? 0.f : f8; a2[1] = hi ? 0.f : f9;

    // B fragment row base per K-chunk (lane-varying address, no branches).
    const int krow = hi ? 2 : 0;             // +0 or +2 rows within chunk

    v8f c[4];
#pragma unroll
    for (int nt = 0; nt < 4; ++nt) {
        const int n = nt * 16 + mrow;
        v2f b0, b1, b2;
        b0[0] = W[(0 + krow) * COUTc + n];
        b0[1] = W[(1 + krow) * COUTc + n];
        b1[0] = W[(4 + krow) * COUTc + n];
        b1[1] = W[(5 + krow) * COUTc + n];
        b2[0] = W[8 * COUTc + n];            // rows 8/9 for ALL lanes:
        b2[1] = W[9 * COUTc + n];            // kp=1 side is nulled via a2==0
        v8f acc = {};
        acc = __builtin_amdgcn_wmma_f32_16x16x4_f32(
            false, a0, false, b0, (short)0, acc, false, false);
        acc = __builtin_amdgcn_wmma_f32_16x16x4_f32(
            false, a1, false, b1, (short)0, acc, false, false);
        acc = __builtin_amdgcn_wmma_f32_16x16x4_f32(
            false, a2, false, b2, (short)0, acc, false, false);
        c[nt] = acc;
    }

    // affine + relu + max over the wave's 16 points, then atomicMax scatter.
#pragma unroll
    for (int nt = 0; nt < 4; ++nt) {
        const int ch = nt * 16 + mrow;
        const float sc  = sscale[ch];
        const float shv = sscale[COUTc + ch];
        float pm = 0.f;                       // relu floor
#pragma unroll
        for (int j = 0; j < 8; ++j) {
            float v = fmaf(c[nt][j], sc, shv);
            pm = fmaxf(pm, fmaxf(v, 0.f));
        }
        pm = fmaxf(pm, __shfl_xor(pm, 16, 32));   // merge the two 8-row halves
        if (lane < 16) {
            // pm >= 0 and out zero-initialized: int-bit atomicMax == float max
            atomicMax((int*)(out + (size_t)obase + (size_t)ch * NYNX),
                      __float_as_int(pm));
        }
    }
}

// ---------------------------------------------------------------------------
extern "C" void kernel_launch(void* const* d_in, const int* in_sizes, int n_in,
                              void* d_out, int out_size, void* d_ws, size_t ws_size,
                              hipStream_t stream) {
    const float* pillars = (const float*)d_in[0];
    const int*   coords  = (const int*)d_in[1];
    const int*   nump    = (const int*)d_in[2];
    const float* W       = (const float*)d_in[3];
    const float* gamma   = (const float*)d_in[4];
    const float* beta    = (const float*)d_in[5];
    float* out = (float*)d_out;

    float* ws     = (float*)d_ws;
    float* stats  = ws;                               // PP*8 floats
    float* part   = ws + (size_t)PP * 8;              // NBLK2*72 floats
    float* sscale = part + (size_t)NBLK2 * 72;        // 128 floats

    zero_out_k<<<4096, 256, 0, stream>>>(out, (long long)out_size);
    pillar_stats_k<<<NBLK2, 256, 0, stream>>>(pillars, coords, nump, stats, part);
    bn_coeffs_k<<<1, 128, 0, stream>>>(part, W, gamma, beta, sscale);
    pfe_wmma_k<<<PP / 4, 256, 0, stream>>>(pillars, W, stats, sscale, out);
}